// LogicLayer_90580860273055
// MI455X (gfx1250) — compile-verified
//
#include <hip/hip_runtime.h>
#include <math.h>

// B=4, N=64, C=8
typedef float v2f __attribute__((ext_vector_type(2)));
typedef float v4f __attribute__((ext_vector_type(4)));
typedef float v8f __attribute__((ext_vector_type(8)));

// sigmoid with hardware v_rcp_f32 (~1 ulp) instead of full IEEE divide chain
__device__ __forceinline__ float sigmoidf(float x) {
    return __builtin_amdgcn_rcpf(1.0f + __expf(-x));
}

// ---------------------------------------------------------------------------
// Masked min/max of x2 over j with mask (i != j).
// R1[b,i, 2c]=max_j(j==i?0:x2), R1[b,i,2c+1]=min_j(j==i?1:x2). 16 floats/(b,i).
// ---------------------------------------------------------------------------
__global__ void k_reduce_x2(const float* __restrict__ x2, float* __restrict__ R1) {
    int t = blockIdx.x * blockDim.x + threadIdx.x;   // B*N*C = 2048
    if (t >= 4 * 64 * 8) return;
    int c = t & 7, i = (t >> 3) & 63, b = t >> 9;
    const float* p = x2 + (((b * 64 + i) * 64) * 8) + c;
    float mx = 0.0f, mn = 1.0f;   // fill values outside mask per reference
    for (int j = 0; j < 64; ++j) {
        if (j == i) continue;
        float v = p[j * 8];
        mx = fmaxf(mx, v);
        mn = fminf(mn, v);
    }
    float* r = R1 + (b * 64 + i) * 16 + 2 * c;
    r[0] = mx;
    r[1] = mn;
}

// ---------------------------------------------------------------------------
// Masked min/max of x3 over k with mask3 = (i!=j)&(i!=k)&(j!=k).
// R2[b,i,j, 2c]=max, R2[...,2c+1]=min (fills 0 / 1 when masked out).
// ---------------------------------------------------------------------------
__global__ void k_reduce_x3(const float* __restrict__ x3, float* __restrict__ R2) {
    int t = blockIdx.x * blockDim.x + threadIdx.x;   // B*N*N*C = 131072
    if (t >= 4 * 64 * 64 * 8) return;
    int c = t & 7, j = (t >> 3) & 63, i = (t >> 9) & 63, b = t >> 15;
    const float* p = x3 + ((((b * 64 + i) * 64 + j) * 64) * 8) + c;
    float mx = 0.0f, mn = 1.0f;
    if (i != j) {
        for (int k = 0; k < 64; ++k) {
            if (k == i || k == j) continue;
            float v = p[k * 8];
            mx = fmaxf(mx, v);
            mn = fminf(mn, v);
        }
    }
    float* r = R2 + ((b * 64 + i) * 64 + j) * 16 + 2 * c;
    r[0] = mx;
    r[1] = mn;
}

// ---------------------------------------------------------------------------
// o0 = sigmoid([x0, minmax(x1)] @ W0 + b0)   (4x24 @ 24x8 -> scalar, tiny)
// ---------------------------------------------------------------------------
__global__ void k_o0(const float* __restrict__ x0, const float* __restrict__ x1,
                     const float* __restrict__ W0, const float* __restrict__ b0,
                     float* __restrict__ out0) {
    int t = threadIdx.x;
    if (t >= 32) return;
    int b = t >> 3, oc = t & 7;
    float f[24];
#pragma unroll
    for (int c = 0; c < 8; ++c) f[c] = x0[b * 8 + c];
    for (int c = 0; c < 8; ++c) {
        float mx = -3.4e38f, mn = 3.4e38f;
        for (int n = 0; n < 64; ++n) {
            float v = x1[(b * 64 + n) * 8 + c];
            mx = fmaxf(mx, v);
            mn = fminf(mn, v);
        }
        f[8 + 2 * c] = mx;
        f[9 + 2 * c] = mn;
    }
    float s = b0[oc];
#pragma unroll
    for (int k = 0; k < 24; ++k) s += f[k] * W0[k * 8 + oc];
    out0[b * 8 + oc] = sigmoidf(s);
}

// ---------------------------------------------------------------------------
// WMMA helpers. V_WMMA_F32_16X16X4_F32 lane mapping (ISA 7.12.2, wave32):
//  lane l: m = l&15 (A row / D col index n), h = l>>4.
//  A frag (v2f) = { A[m][4c+2h], A[m][4c+2h+1] }
//  B frag (v2f) = { B[4c+2h][n], B[4c+2h+1][n] },  n = l&15
//  D: VGPR v, lanes 0-15 -> row v,   lanes 16-31 -> row v+8;  col = l&15.
// W stored in LDS pair-interleaved: WB[p*32 + n*2 + e] = Wpad[2p+e][n].
// ---------------------------------------------------------------------------

// stage W (K x 8) into LDS as pair-interleaved K x 16 (cols 8..15 zero)
__device__ __forceinline__ void stage_W(float* ldsW, const float* W, int K) {
    for (int idx = threadIdx.x; idx < K * 16; idx += blockDim.x) {
        int e = idx & 1, n = (idx >> 1) & 15, p = idx >> 5;
        int k = 2 * p + e;
        ldsW[idx] = (n < 8) ? W[k * 8 + n] : 0.0f;
    }
}

// ---------------------------------------------------------------------------
// o1: rows (b,i) = 256, K=32. feats = [x0[b](8), x1[b,i](8), R1[b,i](16)]
// ---------------------------------------------------------------------------
__global__ void k_o1(const float* __restrict__ x0, const float* __restrict__ x1,
                     const float* __restrict__ R1, const float* __restrict__ W1,
                     const float* __restrict__ b1v, float* __restrict__ out1) {
    __shared__ __attribute__((aligned(16))) float ldsW[32 * 16];
    __shared__ __attribute__((aligned(16))) float ldsA[8][16 * 32];
    stage_W(ldsW, W1, 32);
    __syncthreads();
    int wave = threadIdx.x >> 5, lane = threadIdx.x & 31;
    int tile = blockIdx.x * 8 + wave;        // 16 tiles total
    int it = tile & 3, b = tile >> 2;
    int i0 = it * 16;
    float* A = &ldsA[wave][0];
    for (int t = lane; t < 128; t += 32) {   // 16 rows * 8 float4
        int m = t >> 3, q = t & 7, i = i0 + m;
        const float* src;
        if (q < 2)      src = x0 + b * 8 + q * 4;
        else if (q < 4) src = x1 + (b * 64 + i) * 8 + (q - 2) * 4;
        else            src = R1 + (b * 64 + i) * 16 + (q - 4) * 4;
        *(v4f*)(A + m * 32 + q * 4) = *(const v4f*)src;
    }
    __syncthreads();
    v8f acc = {};
    int n = lane & 15, h = lane >> 4;
#pragma unroll
    for (int c = 0; c < 8; ++c) {
        int kk = 4 * c + 2 * h;
        v2f a = *(const v2f*)(A + n * 32 + kk);
        v2f w = *(const v2f*)(ldsW + (kk >> 1) * 32 + n * 2);
        acc = __builtin_amdgcn_wmma_f32_16x16x4_f32(false, a, false, w,
                                                    (short)0, acc, false, false);
    }
    int rowBase = b * 64 + i0;
    if (n < 8) {
        float bb = b1v[n];
#pragma unroll
        for (int v = 0; v < 8; ++v)
            out1[(rowBase + v + 8 * h) * 8 + n] = sigmoidf(acc[v] + bb);
    }
}

// ---------------------------------------------------------------------------
// o2: rows (b,i,j) = 16384, K=64.
// feats = [x1[b,i], x2[b,i,j], R2[b,i,j](16), x1[b,j], x2[b,j,i], R2[b,j,i](16)]
// ---------------------------------------------------------------------------
__global__ void k_o2(const float* __restrict__ x1, const float* __restrict__ x2,
                     const float* __restrict__ R2, const float* __restrict__ W2,
                     const float* __restrict__ b2v, float* __restrict__ out2) {
    __shared__ __attribute__((aligned(16))) float ldsW[64 * 16];
    __shared__ __attribute__((aligned(16))) float ldsA[8][16 * 64];
    stage_W(ldsW, W2, 64);
    __syncthreads();
    int wave = threadIdx.x >> 5, lane = threadIdx.x & 31;
    int tile = blockIdx.x * 8 + wave;        // 1024 tiles
    int jt = tile & 3, i = (tile >> 2) & 63, b = tile >> 8;
    int j0 = jt * 16;
    float* A = &ldsA[wave][0];
    for (int t = lane; t < 256; t += 32) {   // 16 rows * 16 float4
        int m = t >> 4, q = t & 15, j = j0 + m;
        const float* src;
        if (q < 2)       src = x1 + (b * 64 + i) * 8 + q * 4;
        else if (q < 4)  src = x2 + ((b * 64 + i) * 64 + j) * 8 + (q - 2) * 4;
        else if (q < 8)  src = R2 + ((b * 64 + i) * 64 + j) * 16 + (q - 4) * 4;
        else if (q < 10) src = x1 + (b * 64 + j) * 8 + (q - 8) * 4;
        else if (q < 12) src = x2 + ((b * 64 + j) * 64 + i) * 8 + (q - 10) * 4;
        else             src = R2 + ((b * 64 + j) * 64 + i) * 16 + (q - 12) * 4;
        *(v4f*)(A + m * 64 + q * 4) = *(const v4f*)src;
    }
    __syncthreads();
    v8f acc = {};
    int n = lane & 15, h = lane >> 4;
#pragma unroll
    for (int c = 0; c < 16; ++c) {
        int kk = 4 * c + 2 * h;
        v2f a = *(const v2f*)(A + n * 64 + kk);
        v2f w = *(const v2f*)(ldsW + (kk >> 1) * 32 + n * 2);
        acc = __builtin_amdgcn_wmma_f32_16x16x4_f32(false, a, false, w,
                                                    (short)0, acc, false, false);
    }
    int rowBase = (b * 64 + i) * 64 + j0;
    if (n < 8) {
        float bb = b2v[n];
#pragma unroll
        for (int v = 0; v < 8; ++v)
            out2[(rowBase + v + 8 * h) * 8 + n] = sigmoidf(acc[v] + bb);
    }
}

// ---------------------------------------------------------------------------
// o3: rows (b,i,j,k) = 1,048,576, K=96.
// Perm order from concat: (i,j,k),(i,k,j),(j,i,k),(k,i,j),(j,k,i),(k,j,i);
// group g contributes [x2[b,a1,a2](8), x3[b,a1,a2,a3](8)].
// Tile staging uses CDNA5 async memory->LDS path (GLOBAL_LOAD_ASYNC_TO_LDS_B128,
// GV mode: 64-bit VGPR address + 'off' (SADDR=NULL); LDS byte offset in vdst
// VGPR; tracked on ASYNCcnt).
// ---------------------------------------------------------------------------
__global__ void k_o3(const float* __restrict__ x2, const float* __restrict__ x3,
                     const float* __restrict__ W3, const float* __restrict__ b3v,
                     float* __restrict__ out3) {
    __shared__ __attribute__((aligned(16))) float ldsW[96 * 16];
    __shared__ __attribute__((aligned(16))) float ldsA[8][16 * 96];
    stage_W(ldsW, W3, 96);
    int wave = threadIdx.x >> 5, lane = threadIdx.x & 31;
    int tile = blockIdx.x * 8 + wave;        // 65536 tiles
    int kt = tile & 3, j = (tile >> 2) & 63, i = (tile >> 8) & 63, b = tile >> 14;
    int k0 = kt * 16;
    float* A = &ldsA[wave][0];
    for (int t = lane; t < 384; t += 32) {   // 16 rows * 24 float4
        int m = t / 24;
        int r = t - m * 24;
        int g = r >> 2;                       // permutation group 0..5
        int q = r & 3;                        // 0,1 -> x2 half ; 2,3 -> x3 half
        int k = k0 + m;
        int a1, a2, a3;
        switch (g) {
            case 0: a1 = i; a2 = j; a3 = k; break;
            case 1: a1 = i; a2 = k; a3 = j; break;
            case 2: a1 = j; a2 = i; a3 = k; break;
            case 3: a1 = k; a2 = i; a3 = j; break;
            case 4: a1 = j; a2 = k; a3 = i; break;
            default: a1 = k; a2 = j; a3 = i; break;
        }
        const float* src;
        if (q < 2)
            src = x2 + ((b * 64 + a1) * 64 + a2) * 8 + q * 4;
        else
            src = x3 + (((b * 64 + a1) * 64 + a2) * 64 + a3) * 8 + (q - 2) * 4;
        // LDS byte offset: low 32 bits of the flat __shared__ address are the
        // workgroup-relative LDS address (aperture maps addr[31:0] -> LDS).
        unsigned lds_off = (unsigned)(uintptr_t)(A + m * 96 + g * 16 + q * 4);
        unsigned long long gaddr = (unsigned long long)(uintptr_t)src;
        asm volatile("global_load_async_to_lds_b128 %0, %1, off"
                     :
                     : "v"(lds_off), "v"(gaddr)
                     : "memory");
    }
    // async loads tracked on ASYNCcnt; drain before cross-lane LDS reads
    asm volatile("s_wait_asynccnt 0x0" ::: "memory");
    __syncthreads();
    v8f acc = {};
    int n = lane & 15, h = lane >> 4;
#pragma unroll
    for (int c = 0; c < 24; ++c) {
        int kk = 4 * c + 2 * h;
        v2f a = *(const v2f*)(A + n * 96 + kk);
        v2f w = *(const v2f*)(ldsW + (kk >> 1) * 32 + n * 2);
        acc = __builtin_amdgcn_wmma_f32_16x16x4_f32(false, a, false, w,
                                                    (short)0, acc, false, false);
    }
    int rowBase = ((b * 64 + i) * 64 + j) * 64 + k0;
    if (n < 8) {
        float bb = b3v[n];
#pragma unroll
        for (int v = 0; v < 8; ++v)
            out3[(rowBase + v + 8 * h) * 8 + n] = sigmoidf(acc[v] + bb);
    }
}

// ---------------------------------------------------------------------------
extern "C" void kernel_launch(void* const* d_in, const int* in_sizes, int n_in,
                              void* d_out, int out_size, void* d_ws, size_t ws_size,
                              hipStream_t stream) {
    (void)in_sizes; (void)n_in; (void)out_size; (void)ws_size;
    const float* x0 = (const float*)d_in[0];
    const float* x1 = (const float*)d_in[1];
    const float* x2 = (const float*)d_in[2];
    const float* x3 = (const float*)d_in[3];
    const float* W0 = (const float*)d_in[4];
    const float* b0 = (const float*)d_in[5];
    const float* W1 = (const float*)d_in[6];
    const float* b1 = (const float*)d_in[7];
    const float* W2 = (const float*)d_in[8];
    const float* b2 = (const float*)d_in[9];
    const float* W3 = (const float*)d_in[10];
    const float* b3 = (const float*)d_in[11];

    float* out = (float*)d_out;
    float* o0 = out;                                   // 4*8
    float* o1 = out + 32;                              // 4*64*8
    float* o2 = out + 32 + 4 * 64 * 8;                 // 4*64*64*8
    float* o3 = out + 32 + 4 * 64 * 8 + 4 * 64 * 64 * 8;

    float* R2 = (float*)d_ws;                          // 4*64*64*16 floats (1 MB)
    float* R1 = R2 + 4 * 64 * 64 * 16;                 // 4*64*16 floats (16 KB)

    hipLaunchKernelGGL(k_reduce_x2, dim3(8),    dim3(256), 0, stream, x2, R1);
    hipLaunchKernelGGL(k_reduce_x3, dim3(512),  dim3(256), 0, stream, x3, R2);
    hipLaunchKernelGGL(k_o0,        dim3(1),    dim3(32),  0, stream, x0, x1, W0, b0, o0);
    hipLaunchKernelGGL(k_o1,        dim3(2),    dim3(256), 0, stream, x0, x1, R1, W1, b1, o1);
    hipLaunchKernelGGL(k_o2,        dim3(128),  dim3(256), 0, stream, x1, x2, R2, W2, b2, o2);
    hipLaunchKernelGGL(k_o3,        dim3(8192), dim3(256), 0, stream, x2, x3, W3, b3, o3);
}